// STPCell_33045478375637
// MI455X (gfx1250) — compile-verified
//
#include <hip/hip_runtime.h>
#include <math.h>

#define H_SZ 1024
#define B_SZ 32
#define I_SZ 512

typedef float v2f __attribute__((ext_vector_type(2)));
typedef float v4f __attribute__((ext_vector_type(4)));
typedef float v8f __attribute__((ext_vector_type(8)));

__device__ __forceinline__ float sigf(float v) {
    return 1.0f / (1.0f + __expf(-v));
}

__device__ __forceinline__ v4f sig4(v4f v) {
    v4f r;
    r.x = sigf(v.x); r.y = sigf(v.y); r.z = sigf(v.z); r.w = sigf(v.w);
    return r;
}

// Streaming (single-use) load: non-temporal hint keeps the 256MB U/X streams
// from evicting the small reused tensors (h_t, px, c-rows) out of L2.
__device__ __forceinline__ v4f ldnt4(const float* p) {
    return __builtin_nontemporal_load((const v4f*)p);
}

// ---------------------------------------------------------------------------
// Kernel 0: h_t [H,B] -> htT [B,H] so the main kernel reads h_t[k,b] for
// contiguous k as coalesced float4 loads.
// ---------------------------------------------------------------------------
__global__ __launch_bounds__(256) void transpose_ht(const float* __restrict__ h,
                                                    float* __restrict__ htT) {
    int idx = blockIdx.x * blockDim.x + threadIdx.x;   // idx = k*B + b (coalesced read)
    if (idx < H_SZ * B_SZ) {
        int k = idx / B_SZ;
        int b = idx - k * B_SZ;
        htT[(size_t)b * H_SZ + k] = h[idx];
    }
}

// ---------------------------------------------------------------------------
// Kernel 1: px = p @ x  (fp32 GEMM, 1024x512 * 512x32) via V_WMMA_F32_16X16X4_F32.
// One 16x16 C tile per wave32; 8 waves/block, 16 blocks = 128 tiles.
// A layout (ISA 7.12.2, 32-bit A 16x4): lanes 0-15 -> K=0,1 ; lanes 16-31 -> K=2,3.
// B/C/D: rows striped across lanes within a VGPR.
// ---------------------------------------------------------------------------
__global__ __launch_bounds__(256) void gemm_px(const float* __restrict__ p,
                                               const float* __restrict__ x,
                                               float* __restrict__ px) {
    const int lane = threadIdx.x & 31;
    const int wave = threadIdx.x >> 5;
    const int tile = blockIdx.x * 8 + wave;   // 0..127
    const int rowTile = tile >> 1;            // 0..63  (H/16)
    const int colTile = tile & 1;             // 0..1   (B/16)
    const int m0 = rowTile * 16;
    const int n0 = colTile * 16;
    const int hl = lane >> 4;                 // lane half selects K pair
    const int mr = lane & 15;

    v8f c = {};
    const float* prow = p + (size_t)(m0 + mr) * I_SZ;
    for (int k0 = 0; k0 < I_SZ; k0 += 4) {
        const int ka = k0 + 2 * hl;
        v2f a, b;
        a.x = prow[ka];
        a.y = prow[ka + 1];
        b.x = x[(size_t)ka * B_SZ + n0 + mr];
        b.y = x[(size_t)(ka + 1) * B_SZ + n0 + mr];
        c = __builtin_amdgcn_wmma_f32_16x16x4_f32(
                /*neg_a=*/false, a, /*neg_b=*/false, b,
                /*c_mod=*/(short)0, c, /*reuse_a=*/false, /*reuse_b=*/false);
    }
#pragma unroll
    for (int r = 0; r < 8; ++r) {
        px[(size_t)(m0 + r + 8 * hl) * B_SZ + n0 + mr] = c[r];
    }
}

// ---------------------------------------------------------------------------
// Kernel 2: fused STP update + plasticity-modulated reduction.
// One block per post-synaptic row j. Thread t owns k = 4t..4t+3.
// Per-row sigmoid-derived coefficients computed once, reused for all 32 batches.
// ---------------------------------------------------------------------------
__global__ __launch_bounds__(256) void stp_main(
    const float* __restrict__ h_t, const float* __restrict__ U,
    const float* __restrict__ X,   const float* __restrict__ c_x,
    const float* __restrict__ c_u, const float* __restrict__ c_U,
    const float* __restrict__ c_h, const float* __restrict__ w,
    const float* __restrict__ bias, const float* __restrict__ px,
    const float* __restrict__ htT,  float* __restrict__ out) {
    __shared__ float part[8 * B_SZ];

    const int j    = blockIdx.x;
    const int t    = threadIdx.x;
    const int lane = t & 31;
    const int wv   = t >> 5;
    const int k4   = t * 4;

    const size_t rowOff = (size_t)j * H_SZ + k4;
    const v4f cu4 = *(const v4f*)(c_u + rowOff);
    const v4f cU4 = *(const v4f*)(c_U + rowOff);
    const v4f cx4 = *(const v4f*)(c_x + rowOff);
    const v4f w4  = *(const v4f*)(w   + rowOff);

    // z in [0.001, 0.1]; Ucap = 0.9*sig(c_U)
    const v4f zu = 0.001f + 0.099f * sig4(cu4);
    const v4f uc = 0.9f * sig4(cU4);
    const v4f zx = 0.001f + 0.099f * sig4(cx4);
    const v4f a1 = uc * zu;       // Ucap*z_u
    const v4f a2 = 1.0f - zu;     // (1-z_u)
    const v4f x0 = zx;            // z_x
    const v4f x1 = 1.0f - zx;     // (1-z_x)

    for (int bb = 0; bb < B_SZ; ++bb) {
        const float hjb = h_t[(size_t)j * B_SZ + bb];          // uniform across block
        const size_t base = ((size_t)bb * H_SZ + (size_t)j) * H_SZ + k4;
        const v4f u  = ldnt4(U + base);                        // single-use stream
        const v4f xx = ldnt4(X + base);                        // single-use stream
        const v4f hk = *(const v4f*)(htT + (size_t)bb * H_SZ + k4);

        // U_new = Ucap*z_u + (1-z_u)*U + Ucap*(1-U)*h
        const v4f un = a1 + a2 * u + uc * (1.0f - u) * hjb;
        // X_new = z_x + (1-z_x)*X - U_new*X*h
        const v4f xn = x0 + x1 * xx - un * xx * hjb;
        // contribution to einsum('bjk,kb->jb')
        const v4f m = w4 * un * xn * hk;

        float acc = m.x + m.y + m.z + m.w;
        // wave32 tree reduction
        acc += __shfl_down(acc, 16, 32);
        acc += __shfl_down(acc, 8, 32);
        acc += __shfl_down(acc, 4, 32);
        acc += __shfl_down(acc, 2, 32);
        acc += __shfl_down(acc, 1, 32);
        if (lane == 0) part[wv * B_SZ + bb] = acc;
    }
    __syncthreads();

    if (t < B_SZ) {
        float s = 0.0f;
#pragma unroll
        for (int i = 0; i < 8; ++i) s += part[i * B_SZ + t];
        const float pre = s + px[(size_t)j * B_SZ + t] + bias[j];
        const float zh = 0.01f + (0.5f - 0.01f) * sigf(c_h[j]);
        const float hprev = h_t[(size_t)j * B_SZ + t];
        out[(size_t)j * B_SZ + t] = (1.0f - zh) * hprev + zh * sigf(pre);
    }
}

// ---------------------------------------------------------------------------
// Launch
// ---------------------------------------------------------------------------
extern "C" void kernel_launch(void* const* d_in, const int* in_sizes, int n_in,
                              void* d_out, int out_size, void* d_ws, size_t ws_size,
                              hipStream_t stream) {
    (void)in_sizes; (void)n_in; (void)out_size; (void)ws_size;
    const float* x   = (const float*)d_in[0];   // [I,B]
    const float* h_t = (const float*)d_in[1];   // [H,B]
    const float* U   = (const float*)d_in[2];   // [B,H,H]
    const float* X   = (const float*)d_in[3];   // [B,H,H]
    const float* c_x = (const float*)d_in[4];   // [H,H]
    const float* c_u = (const float*)d_in[5];   // [H,H]
    const float* c_U = (const float*)d_in[6];   // [H,H]
    const float* c_h = (const float*)d_in[7];   // [H,1]
    const float* w   = (const float*)d_in[8];   // [H,H]
    const float* p   = (const float*)d_in[9];   // [H,I]
    const float* b   = (const float*)d_in[10];  // [H,1]
    float* out = (float*)d_out;                 // [H,B]

    float* px  = (float*)d_ws;                  // H*B floats
    float* htT = px + (size_t)H_SZ * B_SZ;      // B*H floats

    transpose_ht<<<(H_SZ * B_SZ + 255) / 256, 256, 0, stream>>>(h_t, htT);
    gemm_px<<<16, 256, 0, stream>>>(p, x, px);
    stp_main<<<H_SZ, 256, 0, stream>>>(h_t, U, X, c_x, c_u, c_U, c_h, w, b,
                                       px, htT, out);
}